// DeepRare_87875030876594
// MI455X (gfx1250) — compile-verified
//
#include <hip/hip_runtime.h>
#include <math.h>
#include <stdint.h>

// DeepRare pipeline for gfx1250 (MI455X). Bandwidth-bound: no GEMM -> no WMMA.
// CDNA5-specific paths: global_load_async_to_lds_b32 + s_wait_asynccnt
// (double-buffered LDS staging in the channel-gather kernel), global_prefetch_b8
// in the streaming sweeps, wave32-safe LDS tree reductions, B128 streaming loads.

#define THRESH  0.2f
#define OUTW    240
#define OUTN    (OUTW*OUTW)
#define MAXC    512

#if defined(__gfx1250__) && \
    __has_builtin(__builtin_amdgcn_global_load_async_to_lds_b32) && \
    __has_builtin(__builtin_amdgcn_s_wait_asynccnt)
#define HAVE_ASYNC 1
#else
#define HAVE_ASYNC 0
#endif

#if HAVE_ASYNC
__device__ __forceinline__ void dr_async_b32(const float* g, float* l) {
  __builtin_amdgcn_global_load_async_to_lds_b32(
      (__attribute__((address_space(1))) int*)(uintptr_t)g,
      (__attribute__((address_space(3))) int*)(uintptr_t)l,
      0, 0);
}
#endif

__device__ __forceinline__ int dr_gidx(float n) {
  // clamp(trunc(n*6-1), 0, 5)  (trunc toward zero == (int) cast)
  float t = n * 6.0f - 1.0f;
  int g = (int)t;
  g = g < 0 ? 0 : (g > 5 ? 5 : g);
  return g;
}

__device__ __forceinline__ int dr_bin(float n) {
  int bi = (int)floorf(n * (6.0f / 256.0f));   // histc bin, width 256/6
  return bi < 0 ? 0 : (bi > 5 ? 5 : bi);
}

// ---------------------------------------------------------------------------
// K1: per-channel min/max (borders read as 0) + histogram + class counts.
// One block (256 thr) per channel; two B128 sweeps; channel < 1MB stays in L2
// between sweeps. Border pixels are constant (0) -> folded in analytically.
// Per-pixel counts kept in registers (unrolled compare-adds), flushed with 18
// LDS atomics per thread.
// ---------------------------------------------------------------------------
__global__ void k_stats_hist(const float* __restrict__ in, int H, int W,
                             float* __restrict__ ws_min, float* __restrict__ ws_max,
                             unsigned* __restrict__ ws_cnt) {
  const int c   = blockIdx.x;
  const int HW  = H * W;
  const int HW4 = HW >> 2;                    // all layer H*W are multiples of 4
  const float* src = in + (size_t)c * HW;
  const float4* src4 = (const float4*)src;
  __shared__ float    smn[256], smx[256];
  __shared__ unsigned scnt[18];
  const int tid = threadIdx.x;

  // ---- sweep 1: interior min/max; border contributes exactly 0 ----
  float mn = 0.0f, mx = 0.0f;                 // seed with border value
  for (int q = tid; q < HW4; q += 256) {
    if (q + 512 < HW4) __builtin_prefetch(&src4[q + 512], 0, 1); // global_prefetch_b8
    float4 xv = src4[q];
    int p = q << 2;
    int i = p / W, j = p - i * W;
    float xs[4] = {xv.x, xv.y, xv.z, xv.w};
#pragma unroll
    for (int k = 0; k < 4; ++k) {
      int jj = j + k, ii = i;
      if (jj >= W) { jj -= W; ii += 1; }      // W >= 4 -> wraps at most once
      bool border = (ii == 0) | (ii == H - 1) | (jj == 0) | (jj == W - 1);
      if (!border) { mn = fminf(mn, xs[k]); mx = fmaxf(mx, xs[k]); }
    }
  }
  smn[tid] = mn; smx[tid] = mx;
  __syncthreads();
  for (int s = 128; s > 0; s >>= 1) {
    if (tid < s) { smn[tid] = fminf(smn[tid], smn[tid + s]); smx[tid] = fmaxf(smx[tid], smx[tid + s]); }
    __syncthreads();
  }
  if (tid == 0) { ws_min[c] = smn[0]; ws_max[c] = smx[0]; }
  if (tid < 18) scnt[tid] = 0u;
  __syncthreads();

  const float gmn = smn[0];
  const float rng = smx[0] - gmn;
  const float scl = rng > 0.0f ? 256.0f / rng : 0.0f;  // rng==0 -> n==0 everywhere

  // ---- sweep 2: interior classes in registers; border folded analytically ----
  int hcnt[6] = {0,0,0,0,0,0}, gcnt[6] = {0,0,0,0,0,0}, gicnt[6] = {0,0,0,0,0,0};
  for (int q = tid; q < HW4; q += 256) {
    if (q + 512 < HW4) __builtin_prefetch(&src4[q + 512], 0, 1);
    float4 xv = src4[q];
    int p = q << 2;
    int i = p / W, j = p - i * W;
    float xs[4] = {xv.x, xv.y, xv.z, xv.w};
#pragma unroll
    for (int k = 0; k < 4; ++k) {
      int jj = j + k, ii = i;
      if (jj >= W) { jj -= W; ii += 1; }
      bool border = (ii == 0) | (ii == H - 1) | (jj == 0) | (jj == W - 1);
      if (!border) {
        float n = (xs[k] - gmn) * scl;
        int bi = dr_bin(n);
        int g  = dr_gidx(n);
#pragma unroll
        for (int b = 0; b < 6; ++b) {
          hcnt[b]  += (bi == b);
          gcnt[b]  += (g == b);
          gicnt[b] += (g == b);
        }
      }
    }
  }
  if (tid == 0) {
    // all 2(H+W)-4 border pixels carry x=0
    int nb = 2 * (H + W) - 4;
    float n0 = (0.0f - gmn) * scl;
    int bi0 = dr_bin(n0), g0 = dr_gidx(n0);
#pragma unroll
    for (int b = 0; b < 6; ++b) {
      hcnt[b] += (bi0 == b) ? nb : 0;
      gcnt[b] += (g0 == b) ? nb : 0;
    }
  }
#pragma unroll
  for (int b = 0; b < 6; ++b) {
    atomicAdd(&scnt[b],      (unsigned)hcnt[b]);
    atomicAdd(&scnt[6 + b],  (unsigned)gcnt[b]);
    atomicAdd(&scnt[12 + b], (unsigned)gicnt[b]);
  }
  __syncthreads();
  if (tid < 18) ws_cnt[c * 18 + tid] = scnt[tid];
}

// ---------------------------------------------------------------------------
// K2: per-channel scalar math -> 8-float record {lut[6], scale0, bias0} + border value.
// Folds _rarity's normalize/ponder and _apply_rarity's per-channel ponder analytically
// using the 6-value class structure.
// ---------------------------------------------------------------------------
__global__ void k_lut(const float* __restrict__ ws_min, const float* __restrict__ ws_max,
                      const unsigned* __restrict__ ws_cnt,
                      float* __restrict__ rec, float* __restrict__ border_arr,
                      int C, int HW) {
  int c = blockIdx.x * blockDim.x + threadIdx.x;
  if (c >= C) return;
  const float N = (float)HW;
  const unsigned* cc = ws_cnt + c * 18;

  float h[6], cg[6], cgi[6];
  for (int g = 0; g < 6; ++g) {
    h[g]   = -logf((float)cc[g] / N + 1e-4f);
    cg[g]  = (float)cc[6 + g];
    cgi[g] = (float)cc[12 + g];
  }
  // dst = normalize(h[gidx], 0, 1); r = ponder(dst) -> v[g] = Wc * q[g]
  float min1 = 3.4e38f, max1 = -3.4e38f;
  for (int g = 0; g < 6; ++g)
    if (cg[g] > 0.0f) { min1 = fminf(min1, h[g]); max1 = fmaxf(max1, h[g]); }
  float rng1 = max1 - min1;
  float v[6];
  if (rng1 > 0.0f) {
    float mean1 = 0.0f;
    for (int g = 0; g < 6; ++g) { v[g] = (h[g] - min1) / rng1; mean1 += cg[g] * v[g]; }
    mean1 /= N;
    float Wc = 1.0f - mean1; Wc *= Wc;           // dst_norm max == 1
    for (int g = 0; g < 6; ++g) v[g] *= Wc;
  } else {
    for (int g = 0; g < 6; ++g) v[g] = 0.0f;
  }
  // maps_c = (c==0) ? r : zero_border(r); ponder(maps_c) statistics from counts.
  float min2 = 3.4e38f, max2 = -3.4e38f, mean2 = 0.0f;
  if (c == 0) {
    for (int g = 0; g < 6; ++g) {
      if (cg[g] > 0.0f) { min2 = fminf(min2, v[g]); max2 = fmaxf(max2, v[g]); }
      mean2 += cg[g] * v[g];
    }
    mean2 /= N;
  } else {
    float nint = 0.0f;
    for (int g = 0; g < 6; ++g) {
      if (cgi[g] > 0.0f) { min2 = fminf(min2, v[g]); max2 = fmaxf(max2, v[g]); }
      mean2 += cgi[g] * v[g]; nint += cgi[g];
    }
    if (N - nint > 0.0f) { min2 = fminf(min2, 0.0f); max2 = fmaxf(max2, 0.0f); }
    mean2 /= N;
  }
  float rng2 = max2 - min2;

  float mn0 = ws_min[c], rng0 = ws_max[c] - mn0;
  float scale0 = rng0 > 0.0f ? 256.0f / rng0 : 0.0f;
  float bias0  = -mn0 * scale0;                 // n = x*scale0 + bias0

  float lut[6], bord;
  if (rng2 > 0.0f) {
    float w2 = max2 - mean2; w2 *= w2;
    float A  = w2 / rng2;
    for (int g = 0; g < 6; ++g) lut[g] = A * (v[g] - min2);
    if (c == 0) {
      int gb = dr_gidx(bias0);                  // border pixel: x = 0 -> n = bias0
      bord = lut[gb];
    } else {
      bord = A * (0.0f - min2);
    }
  } else {
    for (int g = 0; g < 6; ++g) lut[g] = 0.0f;
    bord = 0.0f;
  }
  float* r = rec + c * 8;
  for (int g = 0; g < 6; ++g) r[g] = lut[g];
  r[6] = scale0; r[7] = bias0;
  border_arr[c] = bord;
}

// ---------------------------------------------------------------------------
// K3: proc(p) = sum_c lut_c[class(x_c(p))]; LUT records staged in LDS; the
// per-channel x gather is double-buffered through LDS with async-to-LDS loads.
// ---------------------------------------------------------------------------
__global__ void k_proc(const float* __restrict__ in, const float* __restrict__ rec,
                       const float* __restrict__ border_arr, int C, int H, int Wd,
                       float* __restrict__ proc) {
  __shared__ float s_rec[MAXC * 8];
#if HAVE_ASYNC
  __shared__ float s_stage[2][256];
#endif
  const int tid = threadIdx.x;
  for (int k = tid; k < C * 8; k += 256) s_rec[k] = rec[k];
  __syncthreads();

  const int HW = H * Wd;
  int p = blockIdx.x * 256 + tid;
  if (p >= HW) return;
  int i = p / Wd, j = p - i * Wd;
  bool border = (i == 0) | (i == H - 1) | (j == 0) | (j == Wd - 1);

  float acc = 0.0f;
  if (border) {
    for (int c = 0; c < C; ++c) acc += border_arr[c];
  } else {
    const float* src = in + p;
#if HAVE_ASYNC
    dr_async_b32(src, &s_stage[0][tid]);
    int c = 0;
    for (; c < C - 1; ++c) {
      dr_async_b32(src + (size_t)(c + 1) * HW, &s_stage[(c + 1) & 1][tid]);
      __builtin_amdgcn_s_wait_asynccnt(1);
      float x = s_stage[c & 1][tid];
      float n = fmaf(x, s_rec[c * 8 + 6], s_rec[c * 8 + 7]);
      acc += s_rec[c * 8 + dr_gidx(n)];
    }
    __builtin_amdgcn_s_wait_asynccnt(0);
    {
      float x = s_stage[c & 1][tid];
      float n = fmaf(x, s_rec[c * 8 + 6], s_rec[c * 8 + 7]);
      acc += s_rec[c * 8 + dr_gidx(n)];
    }
#else
    for (int c = 0; c < C; ++c) {
      if (c + 2 < C) __builtin_prefetch(src + (size_t)(c + 2) * HW, 0, 1);
      float x = src[(size_t)c * HW];
      float n = fmaf(x, s_rec[c * 8 + 6], s_rec[c * 8 + 7]);
      acc += s_rec[c * 8 + dr_gidx(n)];
    }
#endif
  }
  proc[p] = acc;
}

// ---------------------------------------------------------------------------
// K4: single-block min/max of proc map (B128 sweep).
// ---------------------------------------------------------------------------
__global__ void k_minmax(const float* __restrict__ v, int n, float* __restrict__ out2) {
  __shared__ float smn[256], smx[256];
  const int tid = threadIdx.x;
  const int n4 = n >> 2;
  const float4* v4 = (const float4*)v;
  float mn = 3.4e38f, mx = -3.4e38f;
  for (int q = tid; q < n4; q += 256) {
    float4 x = v4[q];
    mn = fminf(mn, fminf(fminf(x.x, x.y), fminf(x.z, x.w)));
    mx = fmaxf(mx, fmaxf(fmaxf(x.x, x.y), fmaxf(x.z, x.w)));
  }
  for (int p = (n4 << 2) + tid; p < n; p += 256) { float x = v[p]; mn = fminf(mn, x); mx = fmaxf(mx, x); }
  smn[tid] = mn; smx[tid] = mx;
  __syncthreads();
  for (int s = 128; s > 0; s >>= 1) {
    if (tid < s) { smn[tid] = fminf(smn[tid], smn[tid + s]); smx[tid] = fmaxf(smx[tid], smx[tid + s]); }
    __syncthreads();
  }
  if (tid == 0) { out2[0] = smn[0]; out2[1] = smx[0]; }
}

// ---------------------------------------------------------------------------
// K5: jax.image.resize('bilinear', antialias) HxW -> 240x240; the source is
// normalize(proc,0,1) thresholded at 0.2, applied per-tap on the fly.
// Triangle kernel, kernel_scale = max(1, in/out), per-dim weight normalization.
// ---------------------------------------------------------------------------
__global__ void k_resize(const float* __restrict__ proc, int H, int W,
                         const float* __restrict__ mm, float* __restrict__ outR) {
  int p = blockIdx.x * 256 + threadIdx.x;
  if (p >= OUTN) return;
  int oy = p / OUTW, ox = p - oy * OUTW;

  float mn  = mm[0];
  float rng = mm[1] - mn;
  float inv = rng > 0.0f ? 1.0f / rng : 0.0f;

  float invscale = (float)H / (float)OUTW;      // H == W
  float ks = invscale > 1.0f ? invscale : 1.0f; // antialias widens kernel on downscale
  float cy = ((float)oy + 0.5f) * invscale - 0.5f;
  float cx = ((float)ox + 0.5f) * invscale - 0.5f;

  int y0 = (int)ceilf(cy - ks);  if (y0 < 0) y0 = 0;
  int y1 = (int)floorf(cy + ks); if (y1 > H - 1) y1 = H - 1;
  int x0 = (int)ceilf(cx - ks);  if (x0 < 0) x0 = 0;
  int x1 = (int)floorf(cx + ks); if (x1 > W - 1) x1 = W - 1;

  float wxt = 0.0f;
  for (int x = x0; x <= x1; ++x) {
    float w = 1.0f - fabsf((float)x - cx) / ks; if (w < 0.0f) w = 0.0f;
    wxt += w;
  }
  float wyt = 0.0f, acc = 0.0f;
  for (int y = y0; y <= y1; ++y) {
    float wy = 1.0f - fabsf((float)y - cy) / ks; if (wy < 0.0f) wy = 0.0f;
    wyt += wy;
    if (wy == 0.0f) continue;
    const float* row = proc + (size_t)y * W;
    float racc = 0.0f;
    for (int x = x0; x <= x1; ++x) {
      float wx = 1.0f - fabsf((float)x - cx) / ks; if (wx < 0.0f) wx = 0.0f;
      float t = (row[x] - mn) * inv;             // normalize(proc, 0, 1)
      if (t < THRESH) t = 0.0f;                  // threshold
      racc = fmaf(wx, t, racc);
    }
    acc = fmaf(wy, racc, acc);
  }
  float den = wyt * wxt;
  outR[p] = den > 0.0f ? acc / den : 0.0f;
}

// ---------------------------------------------------------------------------
// K6: min/max/sum of the resized map (B128 sweep).
// ---------------------------------------------------------------------------
__global__ void k_statR(const float* __restrict__ v, int n, float* __restrict__ out3) {
  __shared__ float smn[256], smx[256], ssm[256];
  const int tid = threadIdx.x;
  const int n4 = n >> 2;
  const float4* v4 = (const float4*)v;
  float mn = 3.4e38f, mx = -3.4e38f, sm = 0.0f;
  for (int q = tid; q < n4; q += 256) {
    float4 x = v4[q];
    mn = fminf(mn, fminf(fminf(x.x, x.y), fminf(x.z, x.w)));
    mx = fmaxf(mx, fmaxf(fmaxf(x.x, x.y), fmaxf(x.z, x.w)));
    sm += (x.x + x.y) + (x.z + x.w);
  }
  for (int p = (n4 << 2) + tid; p < n; p += 256) { float x = v[p]; mn = fminf(mn, x); mx = fmaxf(mx, x); sm += x; }
  smn[tid] = mn; smx[tid] = mx; ssm[tid] = sm;
  __syncthreads();
  for (int s = 128; s > 0; s >>= 1) {
    if (tid < s) {
      smn[tid] = fminf(smn[tid], smn[tid + s]);
      smx[tid] = fmaxf(smx[tid], smx[tid + s]);
      ssm[tid] += ssm[tid + s];
    }
    __syncthreads();
  }
  if (tid == 0) { out3[0] = smn[0]; out3[1] = smx[0]; out3[2] = ssm[0]; }
}

// ---------------------------------------------------------------------------
// K7: group = normalize(ponder(resized), 0, 256) == 256*normalize(resized) gated
// by w=(max-mean)^2 > 0 and range > 0; write into interleaved groups output.
// ---------------------------------------------------------------------------
__global__ void k_group(const float* __restrict__ resized, const float* __restrict__ st,
                        float* __restrict__ dout, int layer) {
  int p = blockIdx.x * 256 + threadIdx.x;
  if (p >= OUTN) return;
  float mn = st[0], mx = st[1];
  float mean = st[2] * (1.0f / (float)OUTN);
  float rng = mx - mn;
  float w = mx - mean; w *= w;
  float g = 0.0f;
  if (rng > 0.0f && w > 0.0f) g = (resized[p] - mn) / rng * 256.0f;
  dout[OUTN + (size_t)p * 5 + layer] = g;
}

// ---------------------------------------------------------------------------
// K8: sum plane = groups.sum(-1)
// ---------------------------------------------------------------------------
__global__ void k_final(float* __restrict__ dout) {
  int p = blockIdx.x * 256 + threadIdx.x;
  if (p >= OUTN) return;
  const float* g = dout + OUTN + (size_t)p * 5;
  dout[p] = g[0] + g[1] + g[2] + g[3] + g[4];
}

// ---------------------------------------------------------------------------
// Host launcher. Workspace layout (floats):
//   [0,512)        min0          [512,1024)    max0
//   [1024,10240)   counts (uint, 18 per channel)
//   [10240,14336)  lut records (8 per channel)
//   [14336,14848)  border values
//   [14848,14850)  proc min/max   [14852,14855) resized min/max/sum
//   [14856, +230400) proc map     [245256, +57600) resized map
// Total ~1.16 MB.
// ---------------------------------------------------------------------------
extern "C" void kernel_launch(void* const* d_in, const int* in_sizes, int n_in,
                              void* d_out, int out_size, void* d_ws, size_t ws_size,
                              hipStream_t stream) {
  (void)in_sizes; (void)n_in; (void)out_size; (void)ws_size;
  static const int Cs[5] = {64, 128, 256, 512, 512};
  static const int Hs[5] = {480, 240, 120, 60, 30};

  float*    ws        = (float*)d_ws;
  float*    ws_min    = ws;
  float*    ws_max    = ws + 512;
  unsigned* ws_cnt    = (unsigned*)(ws + 1024);
  float*    ws_rec    = ws + 10240;
  float*    ws_border = ws + 14336;
  float*    ws_mm     = ws + 14848;   // min3, max3
  float*    ws_stR    = ws + 14852;   // minR, maxR, sumR
  float*    ws_proc   = ws + 14856;
  float*    ws_res    = ws + 245256;
  float*    dout      = (float*)d_out;

  for (int L = 0; L < 5; ++L) {
    const int C = Cs[L], H = Hs[L], W = Hs[L], HW = H * W;
    const float* in = (const float*)d_in[L];

    k_stats_hist<<<dim3(C), dim3(256), 0, stream>>>(in, H, W, ws_min, ws_max, ws_cnt);
    k_lut<<<dim3((C + 63) / 64), dim3(64), 0, stream>>>(ws_min, ws_max, ws_cnt,
                                                        ws_rec, ws_border, C, HW);
    k_proc<<<dim3((HW + 255) / 256), dim3(256), 0, stream>>>(in, ws_rec, ws_border,
                                                             C, H, W, ws_proc);
    k_minmax<<<dim3(1), dim3(256), 0, stream>>>(ws_proc, HW, ws_mm);
    k_resize<<<dim3((OUTN + 255) / 256), dim3(256), 0, stream>>>(ws_proc, H, W, ws_mm, ws_res);
    k_statR<<<dim3(1), dim3(256), 0, stream>>>(ws_res, OUTN, ws_stR);
    k_group<<<dim3((OUTN + 255) / 256), dim3(256), 0, stream>>>(ws_res, ws_stR, dout, L);
  }
  k_final<<<dim3((OUTN + 255) / 256), dim3(256), 0, stream>>>(dout);
}